// SSMLayer_21406117003978
// MI455X (gfx1250) — compile-verified
//
#include <hip/hip_runtime.h>

// SSM layer:  out = C @ scan_lambda( (B*dt) @ x )
// All f32, WMMA 16x16x4 f32 for both GEMMs, wave32 complex prefix-scan for the SSM recurrence.

typedef __attribute__((ext_vector_type(2))) float v2f;
typedef __attribute__((ext_vector_type(8))) float v8f;

#define SEQ_L 8192
#define NCH   256
#define LDS_STRIDE 260   // 256 + 4-bank pad: conflict-free ds_load_b64 fragment reads

// ---------------------------------------------------------------------------
// Kernel 1: per-mode lambda (decay, frequency) and dt.
//   dt        = exp(log_dt)
//   lam_a     = -dt * softplus(log_A_real)   (Re log lambda)
//   lam_w     =  dt * A_imag                 (Im log lambda)
// ---------------------------------------------------------------------------
__global__ void ssm_precompute(const float* __restrict__ A,
                               const float* __restrict__ log_dt,
                               float* __restrict__ lam_a,
                               float* __restrict__ lam_w,
                               float* __restrict__ dtv) {
  int n = threadIdx.x;                       // 256 threads, 1 block
  float dt = expf(log_dt[n]);
  float sp = log1pf(expf(A[2 * n]));         // softplus(log_A_real)
  lam_a[n] = -dt * sp;
  lam_w[n] = dt * A[2 * n + 1];
  dtv[n]   = dt;
}

// ---------------------------------------------------------------------------
// Kernel 2/4: dst[b, row, l] = sum_c (Amat[row, c] * scale[row]) * src[b, c, l]
// Amat: 256x256 row-major.  src/dst: [B][256][8192].
// Block = 256 threads (8 waves). Block tile = 64 rows x 32 cols.
// Each wave -> one 16x16 f32 tile via 64x V_WMMA_F32_16X16X4_F32.
// ---------------------------------------------------------------------------
__global__ void __launch_bounds__(256)
ssm_gemm(const float* __restrict__ Amat,
         const float* __restrict__ rowscale, int useScale,
         const float* __restrict__ src,
         float* __restrict__ dst) {
  extern __shared__ float lds[];
  float* lds_a = lds;                    // [64][LDS_STRIDE]  A tile (scaled)
  float* lds_x = lds + 64 * LDS_STRIDE;  // [32][LDS_STRIDE]  src tile, transposed [col][k]

  const int tid  = threadIdx.x;
  const int lane = tid & 31;
  const int wv   = tid >> 5;
  const int wn   = wv & 3;               // wave row-tile   (4 x 16 rows)
  const int wl   = wv >> 2;              // wave col-tile   (2 x 16 cols)
  const int col0 = blockIdx.x * 32;      // L tile
  const int row0 = blockIdx.y * 64;      // output-channel tile
  const size_t bOff = (size_t)blockIdx.z * NCH * SEQ_L;

  // --- stage A tile [64 x 256], float4 coalesced, optional per-row scale ---
  #pragma unroll 4
  for (int it = 0; it < 16; ++it) {
    int idx = it * 1024 + tid * 4;
    int r = idx >> 8;                    // 0..63
    int c = idx & 255;
    float4 v = *(const float4*)(Amat + (size_t)(row0 + r) * NCH + c);
    float s = useScale ? rowscale[row0 + r] : 1.0f;
    float4 sv = make_float4(v.x * s, v.y * s, v.z * s, v.w * s);
    *(float4*)(lds_a + r * LDS_STRIDE + c) = sv;
  }

  // --- stage src tile [256 k x 32 cols] transposed into lds_x[col][k] ---
  {
    int c  = tid & 31;
    int kb = tid >> 5;
    #pragma unroll 8
    for (int kk = 0; kk < 32; ++kk) {
      int k = kb + kk * 8;
      lds_x[c * LDS_STRIDE + k] = src[bOff + (size_t)k * SEQ_L + col0 + c];
    }
  }
  __syncthreads();

  // Fragment addressing per ISA 7.12.2 (f32 16x4 A): lane = M (0..15),
  // VGPR v holds K = v + 2*(lane>=16). Symmetric layout for B (lane = N).
  const int m  = lane & 15;
  const int hi = lane >> 4;
  const float* ap = lds_a + (wn * 16 + m) * LDS_STRIDE + 2 * hi;
  const float* bp = lds_x + (wl * 16 + m) * LDS_STRIDE + 2 * hi;

  v8f acc = {};
  #pragma unroll 8
  for (int c = 0; c < NCH; c += 4) {
    v2f a = *(const v2f*)(ap + c);
    v2f b = *(const v2f*)(bp + c);
    // (neg_a, A, neg_b, B, c_mod, C, reuse_a, reuse_b)
    acc = __builtin_amdgcn_wmma_f32_16x16x4_f32(false, a, false, b,
                                                (short)0, acc, false, false);
  }

  // C/D layout: lanes 0-15: VGPR r -> M=r; lanes 16-31: VGPR r -> M=r+8.
  float* o = dst + bOff + (size_t)(row0 + wn * 16 + 8 * hi) * SEQ_L
                 + col0 + wl * 16 + m;
  #pragma unroll
  for (int r = 0; r < 8; ++r) o[(size_t)r * SEQ_L] = acc[r];
}

// ---------------------------------------------------------------------------
// Kernel 3: in-place causal SSM scan along L.
//   s_l = lambda_n * s_{l-1} + xb_l ,  y_l = Re(s_l)
// One wave32 per (b,n) sequence. 32-element chunks: Hillis-Steele inclusive
// scan with precomputed lambda^(2^s) powers, then carry injection lambda^(k+1).
// ---------------------------------------------------------------------------
__global__ void __launch_bounds__(256)
ssm_scan(float* __restrict__ xb,
         const float* __restrict__ lam_a,
         const float* __restrict__ lam_w) {
  const int lane = threadIdx.x & 31;
  const int wid  = blockIdx.x * 8 + (threadIdx.x >> 5);  // 0..2047 = b*256 + n
  const int n    = wid & 255;
  const size_t base = (size_t)wid * SEQ_L;

  const float a = lam_a[n];
  const float w = lam_w[n];
  const float p = (float)(lane + 1);
  const float er = expf(a * p);
  const float kr = er * cosf(w * p);     // Re(lambda^(lane+1))
  const float ki = er * sinf(w * p);     // Im(lambda^(lane+1))

  float Ar[5], Ai[5];                    // lambda^(2^s): held by lane (2^s - 1)
  #pragma unroll
  for (int s = 0; s < 5; ++s) {
    int d = 1 << s;
    Ar[s] = __shfl(kr, d - 1, 32);
    Ai[s] = __shfl(ki, d - 1, 32);
  }

  float cr = 0.f, ci = 0.f;              // complex carry state
  for (int ch = 0; ch < SEQ_L / 32; ++ch) {
    float pr = xb[base + ch * 32 + lane];
    float pi = 0.f;
    #pragma unroll
    for (int s = 0; s < 5; ++s) {
      int d = 1 << s;
      float hr = __shfl_up(pr, d, 32);
      float hx = __shfl_up(pi, d, 32);
      if (lane >= d) {
        pr += Ar[s] * hr - Ai[s] * hx;
        pi += Ar[s] * hx + Ai[s] * hr;
      }
    }
    float yr = pr + kr * cr - ki * ci;   // inject lambda^(k+1) * carry
    float yi = pi + kr * ci + ki * cr;
    xb[base + ch * 32 + lane] = yr;      // y = Re(s), in place
    cr = __shfl(yr, 31, 32);
    ci = __shfl(yi, 31, 32);
  }
}

// ---------------------------------------------------------------------------
extern "C" void kernel_launch(void* const* d_in, const int* in_sizes, int n_in,
                              void* d_out, int out_size, void* d_ws, size_t ws_size,
                              hipStream_t stream) {
  const float* x      = (const float*)d_in[0];   // (8, 256, 8192)
  const float* A      = (const float*)d_in[1];   // (256, 2)
  const float* B      = (const float*)d_in[2];   // (256, 256)
  const float* log_dt = (const float*)d_in[3];   // (256,)
  const float* C      = (const float*)d_in[4];   // (256, 256)
  float* out = (float*)d_out;                    // (8, 256, 8192)

  float* wsf   = (float*)d_ws;
  float* lam_a = wsf;            // 256
  float* lam_w = wsf + 256;      // 256
  float* dtbuf = wsf + 512;      // 256
  float* xb    = wsf + 1024;     // 8*256*8192 f32 (~64 MB), reused in place for y

  ssm_precompute<<<dim3(1), dim3(256), 0, stream>>>(A, log_dt, lam_a, lam_w, dtbuf);

  dim3 gGemm(SEQ_L / 32, NCH / 64, 8);   // (L tiles, row tiles, batch)
  dim3 blk(256);
  size_t shmem = (size_t)(64 + 32) * LDS_STRIDE * sizeof(float);  // 99,840 B

  // xb[b,n,l] = sum_c (B[n,c]*dt[n]) * x[b,c,l]
  ssm_gemm<<<gGemm, blk, shmem, stream>>>(B, dtbuf, 1, x, xb);

  // y = causal conv with Re(lambda^l)  ==  linear recurrence scan (in place)
  ssm_scan<<<dim3(2048 / 8), blk, 0, stream>>>(xb, lam_a, lam_w);

  // out[b,d,l] = sum_n C[d,n] * y[b,n,l]
  ssm_gemm<<<gGemm, blk, shmem, stream>>>(C, dtbuf, 0, xb, out);
}